// resBlockAutoRegressiveDynamicFLOW_normalDraws_20194936225848
// MI455X (gfx1250) — compile-verified
//
#include <hip/hip_runtime.h>

typedef __attribute__((ext_vector_type(2))) float v2f;
typedef __attribute__((ext_vector_type(8))) float v8f;

#define BROWS   262144
#define HDIM    64
#define DIN     20
#define NCOMP   5
#define OUTC    17      // NC*3+2
#define NITERS  150
#define BNEPS   1e-5f
#define LBV     -20.0f
#define UBV     20.0f
#define PAIRSTRIDE 160  // floats per k-pair row in LDS: 128 data + 32 pad
                        // (half=1 rows start at bank 32 -> conflict-free b64 loads)

// ---------------------------------------------------------------------------
// Zero the 7 stages of column statistics (sum[64], sumsq[64] each)
// ---------------------------------------------------------------------------
__global__ void zero_stats_kernel(float* stats) {
    for (int j = threadIdx.x; j < 7 * 128; j += 256) stats[j] = 0.0f;
}

// ---------------------------------------------------------------------------
// Column statistics over a B x 64 buffer (optionally of relu(x)).
// Grid-stride partial sums -> LDS reduce -> global atomicAdd.
// ---------------------------------------------------------------------------
template<bool RELU>
__global__ __launch_bounds__(256)
void colstats_kernel(const float* __restrict__ src, float* __restrict__ stats) {
    __shared__ float lsum[256];
    __shared__ float lsq[256];
    const int tid = threadIdx.x;
    const int col = tid & 63;
    float s = 0.0f, sq = 0.0f;
    for (int row = blockIdx.x * 4 + (tid >> 6); row < BROWS; row += gridDim.x * 4) {
        float v = src[(size_t)row * HDIM + col];
        if (RELU) v = fmaxf(v, 0.0f);
        s  += v;
        sq += v * v;
    }
    lsum[tid] = s;
    lsq[tid]  = sq;
    __syncthreads();
    if (tid < 64) {
        s  = lsum[tid] + lsum[tid + 64] + lsum[tid + 128] + lsum[tid + 192];
        sq = lsq[tid]  + lsq[tid + 64]  + lsq[tid + 128]  + lsq[tid + 192];
        atomicAdd(&stats[col],      s);
        atomicAdd(&stats[64 + col], sq);
    }
}

// ---------------------------------------------------------------------------
// WMMA fp32 GEMM stage:  dst(B x 64) = pre(src)(B x K) @ W(K x 64) + bias
//   pre(x) = PRE_BN ? s[k]*relu(x)+t[k] : x      (BN folded to per-col affine)
//   RESID: dst += resid (element-aligned with this thread's writes).
// One wave -> one 16x64 output tile via V_WMMA_F32_16X16X4_F32.
// W staged in LDS k-pair interleaved: Wlds[(k>>1)*PAIRSTRIDE + n*2 + (k&1)]
// so each B operand {W[k][n], W[k+1][n]} is one contiguous ds_load_b64.
// ---------------------------------------------------------------------------
template<int KCH, bool PRE_BN, bool RESID>
__global__ __launch_bounds__(256)
void gemm_wmma_kernel(const float* __restrict__ src, int srcStride,
                      const float* __restrict__ W,  const float* __restrict__ bias,
                      const float* __restrict__ stats,
                      const float* __restrict__ g,  const float* __restrict__ be,
                      const float* resid,
                      float* dst) {
    constexpr int K = KCH * 4;
    __shared__ __align__(16) float Wlds[(K / 2) * PAIRSTRIDE];
    __shared__ float blds[HDIM];
    __shared__ float slds[HDIM];
    __shared__ float tlds[HDIM];

    const int tid = threadIdx.x;
    // Coalesced global read of W, scatter into pair-interleaved LDS layout.
    for (int i = tid; i < K * HDIM; i += 256) {
        const int k = i >> 6;          // HDIM == 64
        const int n = i & 63;
        Wlds[(k >> 1) * PAIRSTRIDE + n * 2 + (k & 1)] = W[i];
    }
    if (tid < HDIM) {
        blds[tid] = bias[tid];
        if (PRE_BN) {
            const float invB = 1.0f / (float)BROWS;
            float mu  = stats[tid] * invB;
            float var = stats[64 + tid] * invB - mu * mu;
            float sc  = g[tid] * rsqrtf(var + BNEPS);
            slds[tid] = sc;
            tlds[tid] = be[tid] - mu * sc;
        }
    }
    __syncthreads();

    const int wave    = tid >> 5;
    const int lane    = tid & 31;
    const int half    = lane >> 4;   // 0: K pair {0,1}; 1: K pair {2,3} of each chunk
    const int lr      = lane & 15;
    const int tileRow = (blockIdx.x * 8 + wave) * 16;
    const int row     = tileRow + lr;

    v8f acc[4] = { v8f{}, v8f{}, v8f{}, v8f{} };

    const float* srow = src + (size_t)row * srcStride;
    #pragma unroll
    for (int kc = 0; kc < KCH; ++kc) {
        const int k0 = kc * 4 + half * 2;      // always even
        v2f a = *(const v2f*)(srow + k0);      // single 8B global load
        if (PRE_BN) {
            a.x = fmaxf(a.x, 0.0f) * slds[k0]     + tlds[k0];
            a.y = fmaxf(a.y, 0.0f) * slds[k0 + 1] + tlds[k0 + 1];
        }
        const int prow = kc * 2 + half;        // k-pair row index == k0>>1
        const float* pbase = &Wlds[prow * PAIRSTRIDE + lr * 2];
        #pragma unroll
        for (int nb = 0; nb < 4; ++nb) {
            v2f b = *(const v2f*)(pbase + nb * 32);   // one ds_load_b64
            acc[nb] = __builtin_amdgcn_wmma_f32_16x16x4_f32(
                false, a, false, b, (short)0, acc[nb], false, false);
        }
    }

    // Epilogue: bias (+ residual) and store. C layout: elem r -> row r+8*half, col lr.
    #pragma unroll
    for (int nb = 0; nb < 4; ++nb) {
        const int col = nb * 16 + lr;
        const float bs = blds[col];
        #pragma unroll
        for (int r = 0; r < 8; ++r) {
            const int orow = tileRow + half * 8 + r;
            const size_t idx = (size_t)orow * HDIM + col;
            float v = acc[nb][r] + bs;
            if (RESID) v += resid[idx];
            dst[idx] = v;
        }
    }
}

// ---------------------------------------------------------------------------
// Fused tail: final BN -> relu -> @Wout(64x17)+bout -> mixture params ->
// 150-iteration bisection of mixture CDF. One thread per row.
// ---------------------------------------------------------------------------
__global__ __launch_bounds__(256)
void final_bisect_kernel(const float* __restrict__ h,
                         const float* __restrict__ stats,
                         const float* __restrict__ gf, const float* __restrict__ bf,
                         const float* __restrict__ Wout, const float* __restrict__ bout,
                         const float* __restrict__ u,
                         const float* __restrict__ rescale_w,
                         const float* __restrict__ rescaleAs,
                         float* __restrict__ out) {
    __shared__ float Wl[HDIM * OUTC];
    __shared__ float bl[OUTC];
    __shared__ float sl[HDIM];
    __shared__ float tl[HDIM];

    const int tid = threadIdx.x;
    for (int i = tid; i < HDIM * OUTC; i += 256) Wl[i] = Wout[i];
    if (tid < OUTC) bl[tid] = bout[tid];
    if (tid < HDIM) {
        const float invB = 1.0f / (float)BROWS;
        float mu  = stats[tid] * invB;
        float var = stats[64 + tid] * invB - mu * mu;
        float sc  = gf[tid] * rsqrtf(var + BNEPS);
        sl[tid] = sc;
        tl[tid] = bf[tid] - mu * sc;
    }
    __syncthreads();

    const int row = blockIdx.x * 256 + tid;

    float o[OUTC];
    #pragma unroll
    for (int k = 0; k < OUTC; ++k) o[k] = bl[k];

    const float* hr = h + (size_t)row * HDIM;
    for (int j = 0; j < HDIM; ++j) {
        float hn = fmaxf(sl[j] * hr[j] + tl[j], 0.0f);
        #pragma unroll
        for (int k = 0; k < OUTC; ++k) o[k] += hn * Wl[j * OUTC + k];
    }

    // Mixture parameters
    const float rw  = rescale_w[0];
    const float rAs = rescaleAs[0];

    float loc[NCOMP], wk2[NCOMP], inv[NCOMP];
    float m = o[2 * NCOMP];
    #pragma unroll
    for (int k = 1; k < NCOMP; ++k) m = fmaxf(m, o[2 * NCOMP + k]);
    float wsum = 0.0f;
    #pragma unroll
    for (int k = 0; k < NCOMP; ++k) {
        float e = __expf(o[2 * NCOMP + k] - m);
        wk2[k] = e;
        wsum += e;
    }
    const float winv = 0.5f / wsum;   // fold the 0.5*(1+erf) factor into weights
    #pragma unroll
    for (int k = 0; k < NCOMP; ++k) {
        wk2[k] *= winv;
        loc[k]  = o[k];
        // 1/(std*sqrt(2)) = exp(-rw*tanh(log_scale)) * (1/sqrt(2))
        inv[k]  = __expf(-rw * tanhf(o[NCOMP + k])) * 0.70710678118654752f;
    }
    const float As  = __expf(tanhf(o[3 * NCOMP]) * rAs);
    const float Bs  = o[3 * NCOMP + 1];
    const float obj = 1.0f / (1.0f + __expf(-(u[row] - Bs) / As));

    float a = LBV, b = UBV, c = 0.0f;
    for (int it = 0; it < NITERS; ++it) {
        c = (a + b) * 0.5f;
        float val = -obj;
        #pragma unroll
        for (int k = 0; k < NCOMP; ++k)
            val += wk2[k] * (1.0f + erff((c - loc[k]) * inv[k]));
        if (val > 0.0f) b = c; else a = c;   // gt -> keep a, shrink b
    }
    out[row] = c;
}

// ---------------------------------------------------------------------------
// Launch pipeline
// ---------------------------------------------------------------------------
extern "C" void kernel_launch(void* const* d_in, const int* in_sizes, int n_in,
                              void* d_out, int out_size, void* d_ws, size_t ws_size,
                              hipStream_t stream) {
    (void)in_sizes; (void)n_in; (void)out_size; (void)ws_size;

    // setup_inputs() order: x, u, params{W0,b0, blocks[3]{g1,be1,W1,b1,g2,be2,W2,b2},
    //                                 gf,bf,Wout,bout,rescale_w,rescaleAs}, finalCol
    const float* x  = (const float*)d_in[0];
    const float* u  = (const float*)d_in[1];
    const float* W0 = (const float*)d_in[2];
    const float* b0 = (const float*)d_in[3];
    const float *g1[3], *be1[3], *W1[3], *b1[3], *g2[3], *be2[3], *W2[3], *b2[3];
    for (int i = 0; i < 3; ++i) {
        void* const* p = d_in + 4 + i * 8;
        g1[i]  = (const float*)p[0];
        be1[i] = (const float*)p[1];
        W1[i]  = (const float*)p[2];
        b1[i]  = (const float*)p[3];
        g2[i]  = (const float*)p[4];
        be2[i] = (const float*)p[5];
        W2[i]  = (const float*)p[6];
        b2[i]  = (const float*)p[7];
    }
    const float* gf        = (const float*)d_in[28];
    const float* bf        = (const float*)d_in[29];
    const float* Wout      = (const float*)d_in[30];
    const float* bout      = (const float*)d_in[31];
    const float* rescale_w = (const float*)d_in[32];
    const float* rescaleAs = (const float*)d_in[33];

    float* h     = (float*)d_ws;
    float* r     = h + (size_t)BROWS * HDIM;
    float* stats = r + (size_t)BROWS * HDIM;

    const int gemmGrid = BROWS / 128;   // 8 waves/block * 16 rows/wave
    const int redGrid  = 1024;

    zero_stats_kernel<<<1, 256, 0, stream>>>(stats);

    // h = x @ W0 + b0   (K = 20 -> 5 chunks of 4)
    gemm_wmma_kernel<5, false, false><<<gemmGrid, 256, 0, stream>>>(
        x, DIN, W0, b0, nullptr, nullptr, nullptr, nullptr, h);

    for (int i = 0; i < 3; ++i) {
        float* s1 = stats + (size_t)(2 * i) * 128;
        float* s2 = stats + (size_t)(2 * i + 1) * 128;
        // stats over relu(h)
        colstats_kernel<true><<<redGrid, 256, 0, stream>>>(h, s1);
        // r = bn1(relu(h)) @ W1 + b1
        gemm_wmma_kernel<16, true, false><<<gemmGrid, 256, 0, stream>>>(
            h, HDIM, W1[i], b1[i], s1, g1[i], be1[i], nullptr, r);
        // stats over relu(r)
        colstats_kernel<true><<<redGrid, 256, 0, stream>>>(r, s2);
        // h = h + bn2(relu(r)) @ W2 + b2
        gemm_wmma_kernel<16, true, true><<<gemmGrid, 256, 0, stream>>>(
            r, HDIM, W2[i], b2[i], s2, g2[i], be2[i], h, h);
    }

    // final BN stats over raw h, then fused tail
    float* sF = stats + (size_t)6 * 128;
    colstats_kernel<false><<<redGrid, 256, 0, stream>>>(h, sF);
    final_bisect_kernel<<<BROWS / 256, 256, 0, stream>>>(
        h, sF, gf, bf, Wout, bout, u, rescale_w, rescaleAs, (float*)d_out);
}